// GINNet_59768764892006
// MI455X (gfx1250) — compile-verified
//
#include <hip/hip_runtime.h>
#include <math.h>

// ---------------- problem constants (match reference) ----------------
#define NN      40000      // nodes
#define EE      640000     // edges
#define DIM     128
#define OUTC    16
#define GG      256        // graphs
#define BN_EPS  1e-5f

typedef __attribute__((ext_vector_type(2))) float v2f;
typedef __attribute__((ext_vector_type(8))) float v8f;

#define NTILES (NN / 16)   // 2500 row tiles
#define WPB    8           // waves per block in the big GEMMs
#define KPAIRS (DIM / 2)   // 64 k-pairs

// ---------------------------------------------------------------------
// generic zero fill
__global__ void k_fill0(float* __restrict__ p, int n) {
    int i = blockIdx.x * blockDim.x + threadIdx.x;
    if (i < n) p[i] = 0.0f;
}

// ---------------------------------------------------------------------
// edge aggregation: agg[dst] += x[src]   (segment_sum over edges)
// one wave of 32 lanes covers one edge's 128 channels as float4 per lane
__global__ void k_edge_agg(const int* __restrict__ src, const int* __restrict__ dst,
                           const float* __restrict__ X, float* __restrict__ AGG) {
    int idx = blockIdx.x * blockDim.x + threadIdx.x;   // EE*32 threads
    int e  = idx >> 5;
    if (e >= EE) return;
    int c4 = (idx & 31) << 2;                          // channel*4
    int s = src[e], d = dst[e];
    const float4 v = *(const float4*)(X + (long)s * DIM + c4);
    float* p = AGG + (long)d * DIM + c4;
    atomicAdd(p + 0, v.x);
    atomicAdd(p + 1, v.y);
    atomicAdd(p + 2, v.z);
    atomicAdd(p + 3, v.w);
}

// ---------------------------------------------------------------------
// node pooling: pooled[batch[n]] += x1[n]
__global__ void k_pool(const int* __restrict__ batch, const float* __restrict__ X1,
                       float* __restrict__ POOL) {
    int idx = blockIdx.x * blockDim.x + threadIdx.x;   // NN*32 threads
    int n = idx >> 5;
    if (n >= NN) return;
    int c4 = (idx & 31) << 2;
    int g = batch[n];
    const float4 v = *(const float4*)(X1 + (long)n * DIM + c4);
    float* p = POOL + (long)g * DIM + c4;
    atomicAdd(p + 0, v.x);
    atomicAdd(p + 1, v.y);
    atomicAdd(p + 2, v.z);
    atomicAdd(p + 3, v.w);
}

// ---------------------------------------------------------------------
// Cooperative stage of W into LDS, pair-interleaved so a B fragment for
// WMMA f32 16x16x4 is one contiguous 8B ds_load_b64:
//   Wlds[(k/2)*DIM + n] = { W[k][n], W[k+1][n] }
__device__ __forceinline__
void stage_W_pairs(float2* __restrict__ Wlds, const float* __restrict__ W, int tid) {
    for (int i = tid; i < KPAIRS * DIM; i += WPB * 32) {
        const int kp = i >> 7;          // pair index
        const int n  = i & (DIM - 1);
        float2 v;
        v.x = W[(2 * kp)     * DIM + n];
        v.y = W[(2 * kp + 1) * DIM + n];
        Wlds[i] = v;
    }
}

// ---------------------------------------------------------------------
// GEMM1: H = (X + AGG) @ W + b1, fused BN column statistics (sum, sumsq).
// Block = 8 waves; W (128x128, 64KB) staged once in LDS per block.
// Each wave computes a 16(M) x 128(N) strip: 8 v8f accumulators, A fragment
// loaded once per k-step and reused across the 8 column tiles.
// WMMA f32 16x16x4 VGPR layouts (ISA 7.12.2):
//   A: lanes 0-15 hold K={k0,k0+1}, lanes 16-31 K={k0+2,k0+3}, row = lane%16
//   B: VGPR0 = row (lane<16 ? k0 : k0+2), VGPR1 = row+1, col = lane%16
//   C/D: vgpr i -> row i + 8*(lane/16), col = lane%16
__global__ __launch_bounds__(WPB * 32)
void k_gemm1_bnstat(const float* __restrict__ X, const float* __restrict__ AGG,
                    const float* __restrict__ W, const float* __restrict__ B1,
                    float* __restrict__ H,
                    float* __restrict__ colsum, float* __restrict__ colsq) {
    __shared__ float2 Wlds[KPAIRS * DIM];      // 64 KB, pair-interleaved
    const int tid = threadIdx.x;
    stage_W_pairs(Wlds, W, tid);
    __syncthreads();

    const int wave = tid >> 5;
    const int lane = tid & 31;
    const int rt = blockIdx.x * WPB + wave;    // row-tile index
    if (rt >= NTILES) return;                  // trailing waves (EXEC stays full per-wave)

    const int rlane = lane & 15;
    const int khalf = lane >> 4;               // 0|1
    const int row0  = rt * 16;

    const float2* xr = (const float2*)(X   + (long)(row0 + rlane) * DIM);
    const float2* ar = (const float2*)(AGG + (long)(row0 + rlane) * DIM);
    const v2f*    Wl = (const v2f*)Wlds;

    v8f acc[8] = {};
    for (int k0 = 0; k0 < DIM; k0 += 4) {
        const int kp = (k0 >> 1) + khalf;      // k-pair index for this half-wave
        const float2 xa = xr[kp];
        const float2 aa = ar[kp];
        v2f a;
        a.x = xa.x + aa.x;
        a.y = xa.y + aa.y;
        const v2f* wrow = Wl + kp * DIM + rlane;
#pragma unroll
        for (int ct = 0; ct < 8; ++ct) {
            v2f b = wrow[ct * 16];             // single ds_load_b64, pair-aligned
            acc[ct] = __builtin_amdgcn_wmma_f32_16x16x4_f32(false, a, false, b,
                                                            (short)0, acc[ct], false, false);
        }
    }

    const int rbase = row0 + khalf * 8;
#pragma unroll
    for (int ct = 0; ct < 8; ++ct) {
        const int col = ct * 16 + rlane;
        const float bias = B1[col];
        float s = 0.f, sq = 0.f;
#pragma unroll
        for (int i = 0; i < 8; ++i) {
            float v = acc[ct][i] + bias;
            H[(long)(rbase + i) * DIM + col] = v;
            s  += v;
            sq += v * v;
        }
        atomicAdd(&colsum[col], s);
        atomicAdd(&colsq [col], sq);
    }
}

// ---------------------------------------------------------------------
// BN finalize: per-channel scale/shift from batch stats (128 threads, 1 block)
__global__ void k_bn_finalize(const float* __restrict__ colsum, const float* __restrict__ colsq,
                              const float* __restrict__ gamma, const float* __restrict__ beta,
                              float* __restrict__ scale, float* __restrict__ shift) {
    int c = threadIdx.x;                 // 0..127
    float mu  = colsum[c] * (1.0f / NN);
    float var = colsq[c] * (1.0f / NN) - mu * mu;   // biased variance
    float sc  = gamma[c] * rsqrtf(var + BN_EPS);
    scale[c] = sc;
    shift[c] = beta[c] - mu * sc;
}

// ---------------------------------------------------------------------
// GEMM2: OUT = relu( relu(bn(H)) @ W2 + b2 ); BN+ReLU fused into A load.
// Same blocking as GEMM1: W2 staged pair-interleaved in LDS.
__global__ __launch_bounds__(WPB * 32)
void k_gemm2_relu(const float* __restrict__ H,
                  const float* __restrict__ scale, const float* __restrict__ shift,
                  const float* __restrict__ W, const float* __restrict__ B2,
                  float* __restrict__ OUT) {
    __shared__ float2 Wlds[KPAIRS * DIM];      // 64 KB
    const int tid = threadIdx.x;
    stage_W_pairs(Wlds, W, tid);
    __syncthreads();

    const int wave = tid >> 5;
    const int lane = tid & 31;
    const int rt = blockIdx.x * WPB + wave;
    if (rt >= NTILES) return;

    const int rlane = lane & 15;
    const int khalf = lane >> 4;
    const int row0  = rt * 16;

    const float2* hr  = (const float2*)(H + (long)(row0 + rlane) * DIM);
    const float2* sc2 = (const float2*)scale;
    const float2* sh2 = (const float2*)shift;
    const v2f*    Wl  = (const v2f*)Wlds;

    v8f acc[8] = {};
    for (int k0 = 0; k0 < DIM; k0 += 4) {
        const int kp = (k0 >> 1) + khalf;
        const float2 hv = hr [kp];
        const float2 sv = sc2[kp];
        const float2 tv = sh2[kp];
        v2f a;
        a.x = fmaxf(hv.x * sv.x + tv.x, 0.0f);
        a.y = fmaxf(hv.y * sv.y + tv.y, 0.0f);
        const v2f* wrow = Wl + kp * DIM + rlane;
#pragma unroll
        for (int ct = 0; ct < 8; ++ct) {
            v2f b = wrow[ct * 16];
            acc[ct] = __builtin_amdgcn_wmma_f32_16x16x4_f32(false, a, false, b,
                                                            (short)0, acc[ct], false, false);
        }
    }

    const int rbase = row0 + khalf * 8;
#pragma unroll
    for (int ct = 0; ct < 8; ++ct) {
        const int col = ct * 16 + rlane;
        const float bias = B2[col];
#pragma unroll
        for (int i = 0; i < 8; ++i)
            OUT[(long)(rbase + i) * DIM + col] = fmaxf(acc[ct][i] + bias, 0.0f);
    }
}

// ---------------------------------------------------------------------
// head GEMM: TMP = relu(POOL @ lin1_W + lin1_b)    [256x128 @ 128x128]
// tiny -> single wave per 16x16 tile is plenty
__global__ __launch_bounds__(32)
void k_head_gemm(const float* __restrict__ A, const float* __restrict__ W,
                 const float* __restrict__ B, float* __restrict__ OUT) {
    const int mt = blockIdx.x, nt = blockIdx.y;      // (16, 8)
    const int lane  = threadIdx.x;
    const int rlane = lane & 15;
    const int khalf = lane >> 4;
    const int row0 = mt * 16, col0 = nt * 16;
    const float2* arow = (const float2*)(A + (long)(row0 + rlane) * DIM);

    v8f acc = {};
    for (int k0 = 0; k0 < DIM; k0 += 4) {
        const int ka = k0 + khalf * 2;
        const float2 av = arow[ka >> 1];
        v2f a; a.x = av.x; a.y = av.y;
        v2f b;
        b.x = W[(long)(ka)     * DIM + col0 + rlane];
        b.y = W[(long)(ka + 1) * DIM + col0 + rlane];
        acc = __builtin_amdgcn_wmma_f32_16x16x4_f32(false, a, false, b,
                                                    (short)0, acc, false, false);
    }
    const float bias = B[col0 + rlane];
    const int rbase = row0 + khalf * 8;
#pragma unroll
    for (int i = 0; i < 8; ++i)
        OUT[(long)(rbase + i) * DIM + col0 + rlane] = fmaxf(acc[i] + bias, 0.0f);
}

// ---------------------------------------------------------------------
// final lin2 + log_softmax over 16 classes; one thread per graph (256 total)
__global__ void k_head_final(const float* __restrict__ TMP, const float* __restrict__ W,
                             const float* __restrict__ B, float* __restrict__ OUT) {
    int g = blockIdx.x * blockDim.x + threadIdx.x;
    if (g >= GG) return;
    float o[OUTC];
#pragma unroll
    for (int j = 0; j < OUTC; ++j) o[j] = B[j];
    const float* t = TMP + (long)g * DIM;
    for (int k = 0; k < DIM; ++k) {
        float tv = t[k];
#pragma unroll
        for (int j = 0; j < OUTC; ++j) o[j] += tv * W[k * OUTC + j];
    }
    float m = o[0];
#pragma unroll
    for (int j = 1; j < OUTC; ++j) m = fmaxf(m, o[j]);
    float se = 0.f;
#pragma unroll
    for (int j = 0; j < OUTC; ++j) se += expf(o[j] - m);
    float lse = m + logf(se);
#pragma unroll
    for (int j = 0; j < OUTC; ++j) OUT[(long)g * OUTC + j] = o[j] - lse;
}

// ---------------------------------------------------------------------
extern "C" void kernel_launch(void* const* d_in, const int* in_sizes, int n_in,
                              void* d_out, int out_size, void* d_ws, size_t ws_size,
                              hipStream_t stream) {
    // ---- inputs (setup_inputs dict order) ----
    const float* x       = (const float*)d_in[0];
    const int*   ei      = (const int*)  d_in[1];   // (2, E)
    const int*   batch   = (const int*)  d_in[2];
    // d_in[3] = num_graphs (scalar)
    const float* W1_0 = (const float*)d_in[4];
    const float* b1_0 = (const float*)d_in[5];
    const float* g_0  = (const float*)d_in[6];
    const float* bt_0 = (const float*)d_in[7];
    const float* W2_0 = (const float*)d_in[8];
    const float* b2_0 = (const float*)d_in[9];
    const float* W1s  = (const float*)d_in[10];     // (4,128,128)
    const float* b1s  = (const float*)d_in[11];
    const float* gs   = (const float*)d_in[12];
    const float* bts  = (const float*)d_in[13];
    const float* W2s  = (const float*)d_in[14];
    const float* b2s  = (const float*)d_in[15];
    const float* lin1W = (const float*)d_in[16];
    const float* lin1b = (const float*)d_in[17];
    const float* lin2W = (const float*)d_in[18];
    const float* lin2b = (const float*)d_in[19];

    const int* src = ei;
    const int* dst = ei + EE;

    // ---- workspace layout (floats) ----
    float* ws = (float*)d_ws;
    const long SZ = (long)NN * DIM;          // 5,120,000
    float* agg    = ws;                      // [N, DIM]
    float* hbuf   = ws + SZ;                 // [N, DIM]
    float* bufA   = ws + 2 * SZ;             // [N, DIM]
    float* bufB   = ws + 3 * SZ;             // [N, DIM]
    float* stats  = ws + 4 * SZ;             // colsum[128] colsq[128] scale[128] shift[128]
    float* colsum = stats, *colsq = stats + DIM;
    float* scale  = stats + 2 * DIM, *shift = stats + 3 * DIM;
    float* pooled = stats + 4 * DIM;         // [G, DIM] = 32768
    float* tmp    = pooled + (long)GG * DIM; // [G, DIM] = 32768

    float* out_logits = (float*)d_out;             // [G, OUTC]
    float* out_x1     = (float*)d_out + GG * OUTC; // [N, DIM]

    const int gemmBlocks = (NTILES + WPB - 1) / WPB;   // 313
    const int fillBlocks = (int)((SZ + 255) / 256);
    const int edgeBlocks = (EE * 32 + 255) / 256;
    const int poolBlocks = (NN * 32 + 255) / 256;

    const float* xin = x;
    for (int layer = 0; layer < 5; ++layer) {
        const float* W1, *b1, *gm, *bt, *W2, *b2;
        if (layer == 0) {
            W1 = W1_0; b1 = b1_0; gm = g_0; bt = bt_0; W2 = W2_0; b2 = b2_0;
        } else {
            int i = layer - 1;
            W1 = W1s + (long)i * DIM * DIM; b1 = b1s + (long)i * DIM;
            gm = gs  + (long)i * DIM;       bt = bts + (long)i * DIM;
            W2 = W2s + (long)i * DIM * DIM; b2 = b2s + (long)i * DIM;
        }
        float* xout = (layer == 4) ? out_x1 : ((layer & 1) ? bufB : bufA);

        // zero aggregation buffer + BN stat accumulators
        k_fill0<<<fillBlocks, 256, 0, stream>>>(agg, (int)SZ);
        k_fill0<<<1, 256, 0, stream>>>(stats, 2 * DIM);

        // agg[dst] += x[src]
        k_edge_agg<<<edgeBlocks, 256, 0, stream>>>(src, dst, xin, agg);

        // H = (x + agg) @ W1 + b1   (+ fused BN column stats)
        k_gemm1_bnstat<<<gemmBlocks, WPB * 32, 0, stream>>>(xin, agg, W1, b1, hbuf,
                                                            colsum, colsq);

        // per-channel scale/shift
        k_bn_finalize<<<1, DIM, 0, stream>>>(colsum, colsq, gm, bt, scale, shift);

        // xout = relu( relu(bn(H)) @ W2 + b2 )
        k_gemm2_relu<<<gemmBlocks, WPB * 32, 0, stream>>>(hbuf, scale, shift, W2, b2, xout);

        xin = xout;
    }

    // graph pooling: pooled[g] = sum_{n: batch[n]==g} x1[n]
    k_fill0<<<(GG * DIM + 255) / 256, 256, 0, stream>>>(pooled, GG * DIM);
    k_pool<<<poolBlocks, 256, 0, stream>>>(batch, out_x1, pooled);

    // head: tmp = relu(pooled @ lin1_W + lin1_b)
    const dim3 hgrid(GG / 16, DIM / 16);     // (16, 8)
    k_head_gemm<<<hgrid, 32, 0, stream>>>(pooled, lin1W, lin1b, tmp);

    // logits = log_softmax(tmp @ lin2_W + lin2_b)
    k_head_final<<<(GG + 255) / 256, 256, 0, stream>>>(tmp, lin2W, lin2b, out_logits);
}